// FMFFN_75453985457413
// MI455X (gfx1250) — compile-verified
//
#include <hip/hip_runtime.h>
#include <hip/hip_bf16.h>

typedef __attribute__((ext_vector_type(2))) float v2f;
typedef __attribute__((ext_vector_type(4))) float f4;
typedef __attribute__((ext_vector_type(8))) float v8f;

#define CIN   64
#define HIDD  256
#define HH    256
#define WWID  256
#define TP    8       // 8x8 pixel tile per workgroup (= 2x2 windows of 4x4)
#define RPSTR 160     // dwords per K-row-pair: 64 px * 2 + 32 pad (bank-conflict-free b64 frags)
#define LYP   68      // padded row stride for y tile

__device__ __forceinline__ float gelu_exact(float v) {
    return 0.5f * v * (1.0f + erff(v * 0.70710678118654752440f));
}

__global__ __launch_bounds__(256) void fmffn_fused(
    const float* __restrict__ x,  const float* __restrict__ w1,
    const float* __restrict__ b1, const float* __restrict__ w2,
    const float* __restrict__ b2, const float* __restrict__ cw,
    float* __restrict__ out)
{
    // K-pair interleaved tiles: element (k,p) lives at [(k>>1)*RPSTR + 2p + (k&1)]
    __shared__ float lx2[32 * RPSTR];   // x tile   (64 ch  x 64 px)
    __shared__ float lh2[32 * RPSTR];   // hidden chunk (64 x 64 px), GELU applied
    __shared__ float ly [CIN * LYP];    // y tile before spectral filter

    const int t    = threadIdx.x;
    const int wave = t >> 5;            // wave32
    const int lane = t & 31;
    const int bz = blockIdx.z;
    const int h0 = blockIdx.y * TP;
    const int w0 = blockIdx.x * TP;

    // ---- stage x tile into pair-interleaved LDS via async global->LDS (ASYNCcnt) ----
    #pragma unroll
    for (int i = 0; i < 16; ++i) {
        int idx = t + i * 256;          // 0..4095 -> (c, p); consecutive lanes = consecutive w
        int c = idx >> 6;
        int p = idx & 63;
        int py = p >> 3, px = p & 7;
        const float* gp = x + (((size_t)(bz * CIN + c) * HH + (h0 + py)) * WWID + w0 + px);
        unsigned ldsoff = (unsigned)(uintptr_t)&lx2[(c >> 1) * RPSTR + p * 2 + (c & 1)];
        asm volatile("global_load_async_to_lds_b32 %0, %1, off"
                     :: "v"(ldsoff), "v"(gp) : "memory");
    }
    asm volatile("s_wait_asynccnt 0x0" ::: "memory");
    __syncthreads();

    // wave -> (M tile, pair of N tiles). 4 M-tiles x 4 N-tiles of 16x16.
    const int mt  = wave & 3;
    const int ntb = (wave >> 2) * 2;
    const int l15 = lane & 15;
    const int kh  = (lane >> 4) << 1;   // A/B frag K offset: lanes 0-15 -> {0,1}, 16-31 -> {2,3}
    const int rhi = (lane >> 4) << 3;   // C/D row offset for upper half-wave
    const int n0  = ntb * 16 + l15;
    const int n1  = (ntb + 1) * 16 + l15;
    const int n0x2 = n0 * 2;
    const int n1x2 = n1 * 2;

    v8f y0 = {}; v8f y1 = {};           // GEMM2 accumulators (persist across K chunks)

    for (int chv = 0; chv < 4; ++chv) {
        // ---- GEMM1 chunk: h[chv*64 + r][p] = sum_c w1[o][c] * x[c][p] ----
        v8f a0 = {}; v8f a1 = {};
        const int orow = chv * 64 + mt * 16 + l15;
        #pragma unroll
        for (int k0 = 0; k0 < 16; ++k0) {
            int rp = k0 * 2 + (kh >> 1);                  // (k0*4 + kh) / 2
            v2f af  = *(const v2f*)&w1[orow * CIN + k0 * 4 + kh];
            v2f bf0 = *(const v2f*)&lx2[rp * RPSTR + n0x2];   // single ds_load_b64
            v2f bf1 = *(const v2f*)&lx2[rp * RPSTR + n1x2];
            a0 = __builtin_amdgcn_wmma_f32_16x16x4_f32(false, af, false, bf0, (short)0, a0, false, false);
            a1 = __builtin_amdgcn_wmma_f32_16x16x4_f32(false, af, false, bf1, (short)0, a1, false, false);
        }
        // bias + exact GELU -> LDS hidden chunk (pair-interleaved)
        #pragma unroll
        for (int i = 0; i < 8; ++i) {
            int r = mt * 16 + i + rhi;
            float bias = b1[chv * 64 + r];
            lh2[(r >> 1) * RPSTR + n0x2 + (r & 1)] = gelu_exact(a0[i] + bias);
            lh2[(r >> 1) * RPSTR + n1x2 + (r & 1)] = gelu_exact(a1[i] + bias);
        }
        __syncthreads();

        // ---- GEMM2 partial: y[c][p] += sum_{o in chunk} w2[c][o] * h[o][p] ----
        const int crow = mt * 16 + l15;
        #pragma unroll
        for (int k0 = 0; k0 < 16; ++k0) {
            int rp = k0 * 2 + (kh >> 1);
            v2f af  = *(const v2f*)&w2[crow * HIDD + chv * 64 + k0 * 4 + kh];
            v2f bf0 = *(const v2f*)&lh2[rp * RPSTR + n0x2];
            v2f bf1 = *(const v2f*)&lh2[rp * RPSTR + n1x2];
            y0 = __builtin_amdgcn_wmma_f32_16x16x4_f32(false, af, false, bf0, (short)0, y0, false, false);
            y1 = __builtin_amdgcn_wmma_f32_16x16x4_f32(false, af, false, bf1, (short)0, y1, false, false);
        }
        __syncthreads();   // lh2 reused next chunk
    }

    // ---- bias2, stage y tile ----
    #pragma unroll
    for (int i = 0; i < 8; ++i) {
        int r = mt * 16 + i + rhi;
        float bias = b2[r];
        ly[r * LYP + n0] = y0[i] + bias;
        ly[r * LYP + n1] = y1[i] + bias;
    }
    __syncthreads();

    // ---- spectral filter: per (channel, 4x4 window). 64 ch * 4 windows = 256 threads ----
    {
        const int c  = t >> 2;
        const int wy = (t >> 1) & 1;
        const int wx = t & 1;

        float z[4][4];
        #pragma unroll
        for (int r = 0; r < 4; ++r)
            #pragma unroll
            for (int s = 0; s < 4; ++s)
                z[r][s] = ly[c * LYP + (wy * 4 + r) * 8 + wx * 4 + s];

        // forward DFT along s (k2): twiddles are {1,-i,-1,i}
        float R0[4], R2[4], R1r[4], R1i[4];
        #pragma unroll
        for (int r = 0; r < 4; ++r) {
            R0[r]  = z[r][0] + z[r][1] + z[r][2] + z[r][3];
            R2[r]  = z[r][0] - z[r][1] + z[r][2] - z[r][3];
            R1r[r] = z[r][0] - z[r][2];
            R1i[r] = z[r][3] - z[r][1];
        }
        // forward DFT along r (k1) -> Z[k1][k2], k2 = 0..2
        float Zr[4][3], Zi[4][3];
        Zr[0][0] = R0[0] + R0[1] + R0[2] + R0[3];  Zi[0][0] = 0.f;
        Zr[1][0] = R0[0] - R0[2];                  Zi[1][0] = R0[3] - R0[1];
        Zr[2][0] = R0[0] - R0[1] + R0[2] - R0[3];  Zi[2][0] = 0.f;
        Zr[3][0] = R0[0] - R0[2];                  Zi[3][0] = R0[1] - R0[3];
        Zr[0][2] = R2[0] + R2[1] + R2[2] + R2[3];  Zi[0][2] = 0.f;
        Zr[1][2] = R2[0] - R2[2];                  Zi[1][2] = R2[3] - R2[1];
        Zr[2][2] = R2[0] - R2[1] + R2[2] - R2[3];  Zi[2][2] = 0.f;
        Zr[3][2] = R2[0] - R2[2];                  Zi[3][2] = R2[1] - R2[3];
        Zr[0][1] = R1r[0] + R1r[1] + R1r[2] + R1r[3];
        Zi[0][1] = R1i[0] + R1i[1] + R1i[2] + R1i[3];
        Zr[1][1] = R1r[0] + R1i[1] - R1r[2] - R1i[3];
        Zi[1][1] = R1i[0] - R1r[1] - R1i[2] + R1r[3];
        Zr[2][1] = R1r[0] - R1r[1] + R1r[2] - R1r[3];
        Zi[2][1] = R1i[0] - R1i[1] + R1i[2] - R1i[3];
        Zr[3][1] = R1r[0] - R1i[1] - R1r[2] + R1i[3];
        Zi[3][1] = R1i[0] + R1r[1] - R1i[2] - R1r[3];

        // multiply by complex weight cweight[k1][k2][c][{re,im}]
        #pragma unroll
        for (int k1 = 0; k1 < 4; ++k1)
            #pragma unroll
            for (int k2 = 0; k2 < 3; ++k2) {
                const float* wp = cw + ((size_t)(k1 * 3 + k2) * CIN + c) * 2;
                float wr = wp[0], wi = wp[1];
                float zr = Zr[k1][k2], zi = Zi[k1][k2];
                Zr[k1][k2] = zr * wr - zi * wi;
                Zi[k1][k2] = zr * wi + zi * wr;
            }

        // inverse complex DFT along k1: V[r][k2] = sum_k1 (+i)^(r*k1) * Z[k1][k2]
        float Vr[4][3], Vi[4][3];
        #pragma unroll
        for (int k2 = 0; k2 < 3; ++k2) {
            Vr[0][k2] = Zr[0][k2] + Zr[1][k2] + Zr[2][k2] + Zr[3][k2];
            Vi[0][k2] = Zi[0][k2] + Zi[1][k2] + Zi[2][k2] + Zi[3][k2];
            Vr[1][k2] = Zr[0][k2] - Zi[1][k2] - Zr[2][k2] + Zi[3][k2];
            Vi[1][k2] = Zi[0][k2] + Zr[1][k2] - Zi[2][k2] - Zr[3][k2];
            Vr[2][k2] = Zr[0][k2] - Zr[1][k2] + Zr[2][k2] - Zr[3][k2];
            Vi[2][k2] = Zi[0][k2] - Zi[1][k2] + Zi[2][k2] - Zi[3][k2];
            Vr[3][k2] = Zr[0][k2] + Zi[1][k2] - Zr[2][k2] - Zi[3][k2];
            Vi[3][k2] = Zi[0][k2] - Zr[1][k2] - Zi[2][k2] + Zr[3][k2];
        }

        // irfft along s (Hermitian half k2=0..2), total ortho scale 1/16
        const float sc = 1.0f / 16.0f;
        #pragma unroll
        for (int r = 0; r < 4; ++r) {
            float p0 = Vr[r][0], p1r = Vr[r][1], p1i = Vi[r][1], p2 = Vr[r][2];
            f4 o;
            o.x = (p0 + p2 + 2.0f * p1r) * sc;
            o.y = (p0 - p2 - 2.0f * p1i) * sc;
            o.z = (p0 + p2 - 2.0f * p1r) * sc;
            o.w = (p0 - p2 + 2.0f * p1i) * sc;
            float* gp = out + (((size_t)(bz * CIN + c) * HH + (h0 + wy * 4 + r)) * WWID
                               + w0 + wx * 4);
            *(f4*)gp = o;
        }
    }
}

extern "C" void kernel_launch(void* const* d_in, const int* in_sizes, int n_in,
                              void* d_out, int out_size, void* d_ws, size_t ws_size,
                              hipStream_t stream) {
    const float* x  = (const float*)d_in[0];
    const float* w1 = (const float*)d_in[1];
    const float* b1 = (const float*)d_in[2];
    const float* w2 = (const float*)d_in[3];
    const float* b2 = (const float*)d_in[4];
    const float* cw = (const float*)d_in[5];
    float* out = (float*)d_out;

    dim3 grid(WWID / TP, HH / TP, 8);   // 32 x 32 x B
    dim3 block(256);                    // 8 wave32
    fmffn_fused<<<grid, block, 0, stream>>>(x, w1, b1, w2, b2, cw, out);
}